// GraphConvolution_66984309948595
// MI455X (gfx1250) — compile-verified
//
#include <hip/hip_runtime.h>

// GraphConvolution for MI455X (gfx1250, wave32)
//   out = sum_k A_k @ (x @ W[:,:,k]) + bias
// Phase 0: repack W [D_IN][D_OUT][K] -> Wt [K][D_IN][D_OUT] (lane-coalesced B)
// Phase 1: f32 WMMA GEMM (V_WMMA_F32_16X16X4_F32), B panel register-resident,
//          one wave sweeps a strip of M-tiles -> support[k] in d_ws.
//          Guard-free specialization when n_nodes % 16 == 0 (true for N=50000).
// Phase 2: bias broadcast into d_out
// Phase 3: per-edge gather + fused 3-kernel scale + atomic scatter-add

typedef __attribute__((ext_vector_type(2))) float v2f;
typedef __attribute__((ext_vector_type(8))) float v8f;

#define D           128   // D_IN == D_OUT
#define KK          3     // kernel count
#define MT_PER_WAVE 8     // 16-row M-tiles swept per wave (B reuse factor)

// ---------------------------------------------------------------------------
// Phase 0: Wt[k][i][n] = W[i][n][k]   (196 KB total; trivial)
// ---------------------------------------------------------------------------
__global__ __launch_bounds__(256) void repack_w(
    const float* __restrict__ w, float* __restrict__ wt, int total /* D*D*KK */)
{
  int idx = blockIdx.x * blockDim.x + threadIdx.x;   // idx = (k*D + i)*D + n
  if (idx >= total) return;
  const int n = idx & (D - 1);
  const int i = (idx >> 7) & (D - 1);
  const int k = idx >> 14;
  wt[idx] = w[((size_t)i * D + n) * KK + k];
}

// ---------------------------------------------------------------------------
// Phase 1: support[k][m][n] = sum_i x[m][i] * Wt[k][i][n]
// One wave owns a (k, n0) column panel: loads the 16x128 B panel into 64 VGPRs
// once, then sweeps MT_PER_WAVE M-tiles, 32 chained v_wmma_f32_16x16x4_f32
// per tile over the K=128 reduction. GUARD=false when M is 16-aligned:
// inner loop is pure global_load_b64 + v_wmma (no cndmask, no store compares).
// ---------------------------------------------------------------------------
template <bool GUARD>
__global__ __launch_bounds__(32) void gemm_support_wmma(
    const float* __restrict__ x,        // [N, 128]
    const float* __restrict__ wt,       // [3, 128, 128] repacked
    float* __restrict__ support,        // [3, N, 128] (workspace)
    int n_nodes)
{
  const int lane = threadIdx.x & 31;
  const int hi   = lane >> 4;          // 0: lanes 0-15, 1: lanes 16-31
  const int lo   = lane & 15;
  const int n0   = blockIdx.y * 16;    // out-channel tile
  const int k    = blockIdx.z;         // kernel index
  const int mt0  = blockIdx.x * MT_PER_WAVE;

  // B-matrix panel: 32 chunks of 4x16 f32. Per-lane layout: N = n0+lo,
  // K rows (4c + 2*hi) and (4c + 2*hi + 1). Coalesced across lanes in Wt.
  const float* wk = wt + (size_t)k * D * D;
  v2f B[32];
  #pragma unroll
  for (int c = 0; c < 32; ++c) {
    const float* wp = wk + (size_t)(4 * c + 2 * hi) * D + (n0 + lo);
    B[c].x = wp[0];
    B[c].y = wp[D];
  }

  float* sup = support + (size_t)k * n_nodes * D;

  for (int t = 0; t < MT_PER_WAVE; ++t) {
    const int m0 = (mt0 + t) * 16;
    if (m0 >= n_nodes) return;         // wave-uniform exit

    // A-matrix 16x4 f32 layout: lane holds row M=lo; VGPR0=K(base),
    // VGPR1=K(base+1), base = 4c + 2*hi.
    const int  am      = m0 + lo;
    const bool arow_ok = !GUARD || (am < n_nodes);
    const float* xrow  = x + (size_t)(arow_ok ? am : 0) * D;

    v8f c = {};
    #pragma unroll
    for (int cc = 0; cc < 32; ++cc) {
      v2f a;
      a.x = xrow[4 * cc + 2 * hi + 0];
      a.y = xrow[4 * cc + 2 * hi + 1];
      if (GUARD) {
        a.x = arow_ok ? a.x : 0.0f;
        a.y = arow_ok ? a.y : 0.0f;
      }
      // 8 args: (neg_a, A, neg_b, B, c_mod, C, reuse_a, reuse_b)
      c = __builtin_amdgcn_wmma_f32_16x16x4_f32(
          false, a, false, B[cc], (short)0, c, false, false);
    }

    // C/D layout: VGPR r -> row (m0 + r + 8*hi), col (n0 + lo)
    #pragma unroll
    for (int r = 0; r < 8; ++r) {
      const int m = m0 + r + 8 * hi;
      if (!GUARD || (m < n_nodes)) sup[(size_t)m * D + n0 + lo] = c[r];
    }
  }
}

// ---------------------------------------------------------------------------
// Phase 2: out[m][d] = bias[d]   (fresh every launch; atomics add on top)
// ---------------------------------------------------------------------------
__global__ __launch_bounds__(256) void bias_init(
    const float* __restrict__ bias, float* __restrict__ out, int total)
{
  int i = blockIdx.x * blockDim.x + threadIdx.x;
  if (i < total) out[i] = bias[i & (D - 1)];
}

// ---------------------------------------------------------------------------
// Phase 3: for edge e (row,col): out[row] += sum_k TT[e][k] * support_k[col]
// One wave per edge; lane owns 4 channels (float4). Gathers are L2-resident
// (support = 76.8 MB < 192 MB L2); scatter via global_atomic_add_f32.
// ---------------------------------------------------------------------------
__global__ __launch_bounds__(256) void edge_scatter(
    const float* __restrict__ TT,      // [E, 3]
    const int*   __restrict__ eidx,    // [2, E]
    const float* __restrict__ support, // [3, N, 128]
    float* __restrict__ out,           // [N, 128]
    int n_edges, int n_nodes)
{
  const int lane = threadIdx.x & 31;
  const int wave = threadIdx.x >> 5;
  const int e    = blockIdx.x * (blockDim.x >> 5) + wave;
  if (e >= n_edges) return;

  const int row = eidx[e];
  const int col = eidx[n_edges + e];

  const float t0 = TT[(size_t)e * KK + 0];
  const float t1 = TT[(size_t)e * KK + 1];
  const float t2 = TT[(size_t)e * KK + 2];

  const size_t nd   = (size_t)n_nodes * D;
  const size_t coff = (size_t)col * D;
  const float4* s0 = (const float4*)(support +          coff);
  const float4* s1 = (const float4*)(support +     nd + coff);
  const float4* s2 = (const float4*)(support + 2 * nd + coff);

  const float4 v0 = s0[lane];
  const float4 v1 = s1[lane];
  const float4 v2 = s2[lane];

  float4 m;
  m.x = t0 * v0.x + t1 * v1.x + t2 * v2.x;
  m.y = t0 * v0.y + t1 * v1.y + t2 * v2.y;
  m.z = t0 * v0.z + t1 * v1.z + t2 * v2.z;
  m.w = t0 * v0.w + t1 * v1.w + t2 * v2.w;

  float* o = out + (size_t)row * D + lane * 4;
  atomicAdd(o + 0, m.x);
  atomicAdd(o + 1, m.y);
  atomicAdd(o + 2, m.z);
  atomicAdd(o + 3, m.w);
}

// ---------------------------------------------------------------------------
extern "C" void kernel_launch(void* const* d_in, const int* in_sizes, int n_in,
                              void* d_out, int out_size, void* d_ws, size_t ws_size,
                              hipStream_t stream) {
  const float* x    = (const float*)d_in[0];   // [N, 128]
  const float* TT   = (const float*)d_in[1];   // [E, 3]
  const float* w    = (const float*)d_in[2];   // [128, 128, 3]
  const float* bias = (const float*)d_in[3];   // [128]
  const int*   eidx = (const int*)d_in[4];     // [2, E]
  float*       out  = (float*)d_out;           // [N, 128]

  const int n_nodes = in_sizes[0] / D;         // 50000
  const int n_edges = in_sizes[1] / KK;        // 800000

  // Workspace layout: [ support: 3*N*128 f32 | Wt: 3*128*128 f32 ]
  float* sup = (float*)d_ws;
  float* wt  = sup + (size_t)KK * n_nodes * D;

  // Phase 0: repack W for coalesced B loads
  const int wtotal = D * D * KK;
  repack_w<<<(wtotal + 255) / 256, 256, 0, stream>>>(w, wt, wtotal);

  // Phase 1: WMMA GEMMs -> support (one wave per (k, n0, m-strip))
  const int m_tiles = (n_nodes + 15) / 16;                 // 3125
  dim3 ggrid((m_tiles + MT_PER_WAVE - 1) / MT_PER_WAVE,    // 391 strips
             D / 16,                                       // 8 n-tiles
             KK);                                          // 3 kernels
  if ((n_nodes & 15) == 0) {
    gemm_support_wmma<false><<<ggrid, 32, 0, stream>>>(x, wt, sup, n_nodes);
  } else {
    gemm_support_wmma<true><<<ggrid, 32, 0, stream>>>(x, wt, sup, n_nodes);
  }

  // Phase 2: out = bias (broadcast)
  const int total = n_nodes * D;
  bias_init<<<(total + 255) / 256, 256, 0, stream>>>(bias, out, total);

  // Phase 3: edge gather/scale/scatter (8 edges per 256-thread block)
  edge_scatter<<<(n_edges + 7) / 8, 256, 0, stream>>>(TT, eidx, sup, out,
                                                      n_edges, n_nodes);
}